// RGCN_25606595019037
// MI455X (gfx1250) — compile-verified
//
#include <hip/hip_runtime.h>

// RGCN (basis decomposition) for MI455X / gfx1250, wave32.
//
// Pipeline per layer (r = 0..15 relations, streamed to keep workspace ~53MB,
// everything L2-resident on the 192MB L2):
//   1) combine:  W[r] = sum_b comp[r,b] * V[b]        (tiny)
//   2) GEMM:     Y = X @ W[r]  via V_WMMA_F32_16X16X4_F32 (exact f32)
//   3) scatter:  for edges with etype==r: h[dst] += Y[src] * norm  (f32 atomics)
//   4) bias (+ReLU for layer 1)

typedef __attribute__((ext_vector_type(2))) float v2f;
typedef __attribute__((ext_vector_type(8))) float v8f;

#define H_DIM   128
#define OUT_DIM 64
#define N_RELS  16

// ---------------------------------------------------------------- combine ---
// W[r, i, o] = sum_b comp[r, b] * V[b, i, o]        (inout_sz = IN*OUT)
__global__ void rgcn_combine(const float* __restrict__ comp,
                             const float* __restrict__ V,
                             float* __restrict__ W, int inout_sz) {
    int idx = blockIdx.x * blockDim.x + threadIdx.x;
    if (idx >= N_RELS * inout_sz) return;
    int r = idx / inout_sz;
    int rem = idx - r * inout_sz;
    float acc = 0.f;
#pragma unroll
    for (int b = 0; b < N_RELS; ++b)
        acc += comp[r * N_RELS + b] * V[(size_t)b * inout_sz + rem];
    W[idx] = acc;
}

// ------------------------------------------------------------- init bias ----
// h[n, c] = bias[c]   (out must be a power of two: 128 / 64)
__global__ void rgcn_init_bias(float* __restrict__ h,
                               const float* __restrict__ bias,
                               int total, int out_mask) {
    int idx = blockIdx.x * blockDim.x + threadIdx.x;
    if (idx >= total) return;
    h[idx] = bias[idx & out_mask];
}

// ---------------------------------------------------------------- GEMM ------
// Y[nrows, OUT] = X[nrows, H_DIM] @ W[H_DIM, OUT], one wave per 16x16 tile.
// Block: 16 output rows x full OUT width; threadIdx.y = column-tile wave.
// f32 WMMA fragment layout (ISA 7.12.2, wave32):
//   A 16x4:  v0 = {lanes0-15: K=k0, lanes16-31: K=k0+2}, v1 = K+1 of those
//   B 4x16:  mirrored (row K striped across lanes, N = lane&15)
//   C/D:     vgpr i -> M = i + 8*(lane>=16), N = lane&15
template <int OUT>
__global__ void rgcn_gemm_wmma(const float* __restrict__ X,
                               const float* __restrict__ W,
                               float* __restrict__ Y) {
    const int lane = threadIdx.x;          // 0..31
    const int m    = lane & 15;
    const int khi  = lane >> 4;            // 0 or 1
    const int row0 = blockIdx.x * 16;
    const int col0 = threadIdx.y * 16;

    const float* xrow = X + (size_t)(row0 + m) * H_DIM;
    v8f c = {};

#pragma unroll
    for (int k0 = 0; k0 < H_DIM; k0 += 4) {
        const int ka = k0 + 2 * khi;
        v2f a, b;
        a.x = xrow[ka];
        a.y = xrow[ka + 1];
        b.x = W[(size_t)ka * OUT + col0 + m];
        b.y = W[(size_t)(ka + 1) * OUT + col0 + m];
        // (neg_a, A, neg_b, B, c_mod, C, reuse_a, reuse_b)
        c = __builtin_amdgcn_wmma_f32_16x16x4_f32(
                false, a, false, b, (short)0, c, false, false);
    }

    float* yout = Y + (size_t)row0 * OUT + col0 + m;
#pragma unroll
    for (int i = 0; i < 8; ++i)
        yout[(size_t)(i + 8 * khi) * OUT] = c[i];
}

// --------------------------------------------------------------- scatter ----
// One wave32 per edge; lanes stride the feature dim. Only edges with
// etype == rel are processed in this pass (Y holds relation rel's transform).
template <int OUT>
__global__ void rgcn_scatter(const float* __restrict__ Y,
                             const int* __restrict__ src,
                             const int* __restrict__ dst,
                             const int* __restrict__ etype,
                             const float* __restrict__ norm,
                             float* __restrict__ h, int rel, int nedges) {
    const int lane = threadIdx.x & 31;
    const int e = blockIdx.x * (blockDim.x >> 5) + (threadIdx.x >> 5);
    if (e >= nedges || etype[e] != rel) return;

    const float w = norm[e];
    const float* ys = Y + (size_t)src[e] * OUT;
    float*       hd = h + (size_t)dst[e] * OUT;
#pragma unroll
    for (int j = 0; j < OUT / 32; ++j) {
        float v = ys[lane + 32 * j] * w;
        __hip_atomic_fetch_add(&hd[lane + 32 * j], v,
                               __ATOMIC_RELAXED, __HIP_MEMORY_SCOPE_AGENT);
    }
}

// ----------------------------------------------------------------- relu -----
__global__ void rgcn_relu(float* __restrict__ h, int n) {
    int i = blockIdx.x * blockDim.x + threadIdx.x;
    if (i < n) h[i] = fmaxf(h[i], 0.f);
}

// ---------------------------------------------------------------- launch ----
extern "C" void kernel_launch(void* const* d_in, const int* in_sizes, int n_in,
                              void* d_out, int out_size, void* d_ws, size_t ws_size,
                              hipStream_t stream) {
    (void)n_in; (void)out_size; (void)ws_size;
    // inputs: 0 node_ids (identity, unused), 1 src, 2 dst, 3 etype, 4 norm,
    //         5 emb, 6 V1, 7 comp1, 8 bias1, 9 V2, 10 comp2, 11 bias2
    const int*   src   = (const int*)d_in[1];
    const int*   dst   = (const int*)d_in[2];
    const int*   etype = (const int*)d_in[3];
    const float* norm  = (const float*)d_in[4];
    const float* emb   = (const float*)d_in[5];
    const float* V1    = (const float*)d_in[6];
    const float* comp1 = (const float*)d_in[7];
    const float* bias1 = (const float*)d_in[8];
    const float* V2    = (const float*)d_in[9];
    const float* comp2 = (const float*)d_in[10];
    const float* bias2 = (const float*)d_in[11];

    const int N = in_sizes[0];   // 50000 (divisible by 16)
    const int E = in_sizes[1];   // 1000000

    // workspace: W1c | W2c | Y (reused per relation) | h1   (~52.8 MB)
    float* W1c = (float*)d_ws;                                   // 16*128*128
    float* W2c = W1c + (size_t)N_RELS * H_DIM * H_DIM;           // 16*128*64
    float* Yb  = W2c + (size_t)N_RELS * H_DIM * OUT_DIM;         // N*128
    float* h1  = Yb + (size_t)N * H_DIM;                         // N*128
    float* out = (float*)d_out;

    // basis combines
    {
        int tot = N_RELS * H_DIM * H_DIM;
        rgcn_combine<<<(tot + 255) / 256, 256, 0, stream>>>(comp1, V1, W1c,
                                                            H_DIM * H_DIM);
    }
    {
        int tot = N_RELS * H_DIM * OUT_DIM;
        rgcn_combine<<<(tot + 255) / 256, 256, 0, stream>>>(comp2, V2, W2c,
                                                            H_DIM * OUT_DIM);
    }

    const int rowTiles    = N / 16;
    const int wavesPerBlk = 8;                       // 256 threads = 8 waves
    const int scatBlocks  = (E + wavesPerBlk - 1) / wavesPerBlk;

    // ---- layer 1: h1 = relu(bias1 + scatter(emb @ W1[r])) ----
    {
        int tot = N * H_DIM;
        rgcn_init_bias<<<(tot + 255) / 256, 256, 0, stream>>>(h1, bias1, tot,
                                                              H_DIM - 1);
    }
    for (int r = 0; r < N_RELS; ++r) {
        rgcn_gemm_wmma<H_DIM>
            <<<dim3(rowTiles), dim3(32, H_DIM / 16), 0, stream>>>(
                emb, W1c + (size_t)r * H_DIM * H_DIM, Yb);
        rgcn_scatter<H_DIM><<<scatBlocks, wavesPerBlk * 32, 0, stream>>>(
            Yb, src, dst, etype, norm, h1, r, E);
    }
    rgcn_relu<<<(N * H_DIM + 255) / 256, 256, 0, stream>>>(h1, N * H_DIM);

    // ---- layer 2: out = bias2 + scatter(h1 @ W2[r]) ----
    {
        int tot = N * OUT_DIM;
        rgcn_init_bias<<<(tot + 255) / 256, 256, 0, stream>>>(out, bias2, tot,
                                                              OUT_DIM - 1);
    }
    for (int r = 0; r < N_RELS; ++r) {
        rgcn_gemm_wmma<OUT_DIM>
            <<<dim3(rowTiles), dim3(32, OUT_DIM / 16), 0, stream>>>(
                h1, W2c + (size_t)r * H_DIM * OUT_DIM, Yb);
        rgcn_scatter<OUT_DIM><<<scatBlocks, wavesPerBlk * 32, 0, stream>>>(
            Yb, src, dst, etype, norm, out, r, E);
    }
}